// _ALIFNet_14894946582853
// MI455X (gfx1250) — compile-verified
//
#include <hip/hip_runtime.h>
#include <stdint.h>

typedef __attribute__((ext_vector_type(2))) float v2f;
typedef __attribute__((ext_vector_type(4))) float v4f;
typedef __attribute__((ext_vector_type(8))) float v8f;

constexpr int   Bn      = 64;
constexpr int   Tn      = 256;
constexpr int   Dn      = 512;
constexpr int   Hn      = 1024;
constexpr int   HTILE   = 128;            // h columns per workgroup
constexpr int   MT      = 16;             // timesteps per chunk (WMMA M)
constexpr int   NCHUNK  = Tn / MT;        // 16
constexpr int   NTILES  = Hn / HTILE;     // 8
constexpr int   LDA     = Dn + 4;         // 516: pad -> conflict-free b64 ds reads
constexpr int   LDP     = HTILE + 2;      // 130: pad -> conflict-free scatter
constexpr float BETA       = 0.9f;
constexpr float ADAPT_RATE = 0.1f;

// Stage one 16 x 512 f32 x-chunk into an LDS buffer via the CDNA5 async
// DMA path (ASYNCcnt-tracked, bypasses VGPRs, does not perturb LOADcnt
// ordering of the W_in loads inside the WMMA loop).
__device__ __forceinline__ void async_stage_chunk(
    const float* __restrict__ gsrc,   // xb + t0*Dn (contiguous 8192 floats)
    float* lbuf,                      // LDS buffer base (row stride LDA)
    int tid)
{
    #pragma unroll
    for (int j = 0; j < 8; ++j) {
        int i   = tid + (j << 8);         // float4 index 0..2047
        int row = i >> 7;                 // / 128
        int col = (i & 127) << 2;         // * 4
        const float* g = gsrc + (i << 2);
        uint32_t l = (uint32_t)(uintptr_t)(lbuf + row * LDA + col);
        asm volatile("global_load_async_to_lds_b128 %0, %1, off"
                     :: "v"(l), "v"(g) : "memory");
    }
}

__device__ __forceinline__ void wait_async_all()
{
    asm volatile("s_wait_asynccnt 0" ::: "memory");
}

__global__ __launch_bounds__(256) void alif_fused(
    const float* __restrict__ x,       // [B,T,D]
    const float* __restrict__ W_in,    // [H,D]
    const float* __restrict__ b_in,    // [H]
    const float* __restrict__ W_out,   // [1,H]
    float* __restrict__ ws)            // [B, NTILES] partials
{
    __shared__ float lA[2][MT * LDA];  // double-buffered x chunk (2 x ~33 KB)
    __shared__ float lP[MT * LDP];     // pre chunk: 16 x 130 floats
    __shared__ float lRed[256];

    const int b    = blockIdx.y;
    const int til  = blockIdx.x;
    const int h0   = til * HTILE;
    const int tid  = threadIdx.x;
    const int lane = tid & 31;
    const int wave = tid >> 5;

    // WMMA f32 16x16x4 fragment coordinates (wave32):
    //   A: lane<16 -> M=lane, K = {0,1}; lane>=16 -> M=lane-16, K = {2,3}
    //   B: lane<16 -> N=lane, K = {0,1}; lane>=16 -> N=lane-16, K = {2,3}
    const int mn    = lane & 15;            // M (time row) for A, N (h col) for B
    const int khalf = (lane >> 4) << 1;     // 0 or 2

    // This wave owns h columns [h0 + wave*16, h0 + wave*16 + 16)
    const float* wrow = W_in + (size_t)(h0 + (wave << 4) + mn) * Dn + khalf;
    const float  bias = b_in[h0 + (wave << 4) + mn];

    // ALIF state: thread t (< 128) owns h = h0 + t
    float mem = 0.0f, adapt = 0.0f, spk_sum = 0.0f;

    const float* xb = x + (size_t)b * Tn * Dn;

    // Prologue: async-stage chunk 0 into buffer 0
    async_stage_chunk(xb, lA[0], tid);
    wait_async_all();
    __syncthreads();

    for (int chunk = 0; chunk < NCHUNK; ++chunk) {
        const int cur = chunk & 1;
        float* curA = lA[cur];

        // Kick off the async DMA for the next chunk into the other buffer;
        // it streams while the WMMA loop below runs.
        if (chunk + 1 < NCHUNK)
            async_stage_chunk(xb + (size_t)(chunk + 1) * MT * Dn,
                              lA[cur ^ 1], tid);

        // 16(t) x 16(h) tile per wave, K = 512 in steps of 4, unroll x2
        v8f acc0 = {};
        v8f acc1 = {};
        const float* arow = &curA[mn * LDA + khalf];
        #pragma unroll 4
        for (int k0 = 0; k0 < Dn; k0 += 8) {
            v2f a0 = *(const v2f*)(arow + k0);
            v2f b0 = *(const v2f*)(wrow + k0);
            v2f a1 = *(const v2f*)(arow + k0 + 4);
            v2f b1 = *(const v2f*)(wrow + k0 + 4);
            acc0 = __builtin_amdgcn_wmma_f32_16x16x4_f32(
                       false, a0, false, b0, (short)0, acc0, false, false);
            acc1 = __builtin_amdgcn_wmma_f32_16x16x4_f32(
                       false, a1, false, b1, (short)0, acc1, false, false);
        }
        v8f acc = acc0 + acc1;

        // Scatter C to lP: element r -> t = r + 8*(lane>=16), col = wave*16 + mn
        {
            const int colp = (wave << 4) + mn;
            const int roff = (lane >> 4) << 3;
            #pragma unroll
            for (int r = 0; r < 8; ++r)
                lP[(r + roff) * LDP + colp] = acc[r] + bias;
        }

        // Drain this wave's async transfers for chunk+1, then rendezvous:
        // after this barrier, lP is visible, curA is fully consumed by all
        // waves, and the next buffer is fully written.
        wait_async_all();
        __syncthreads();

        // ALIF recurrence over this chunk's 16 timesteps (t-serial)
        if (tid < HTILE) {
            #pragma unroll
            for (int tt = 0; tt < MT; ++tt) {
                float p = lP[tt * LDP + tid];
                mem = BETA * mem + p - adapt;
                float spk = (mem > 0.0f) ? 1.0f : 0.0f;
                adapt += ADAPT_RATE * spk;
                mem -= spk;              // reset by subtraction
                spk_sum += spk;
            }
        }
        __syncthreads();
    }

    // Partial of the output dot: sum_h spk_sum[h] * W_out[h]
    lRed[tid] = (tid < HTILE) ? spk_sum * W_out[h0 + tid] : 0.0f;
    __syncthreads();
    #pragma unroll
    for (int s = 128; s > 0; s >>= 1) {
        if (tid < s) lRed[tid] += lRed[tid + s];
        __syncthreads();
    }
    if (tid == 0) ws[b * NTILES + til] = lRed[0];
}

__global__ void alif_finish(const float* __restrict__ ws,
                            const float* __restrict__ b_out,
                            float* __restrict__ out)
{
    int bb = threadIdx.x;
    if (bb < Bn) {
        float s = 0.0f;
        #pragma unroll
        for (int j = 0; j < NTILES; ++j) s += ws[bb * NTILES + j];
        out[bb] = b_out[0] + s * (1.0f / (float)Tn);
    }
}

extern "C" void kernel_launch(void* const* d_in, const int* in_sizes, int n_in,
                              void* d_out, int out_size, void* d_ws, size_t ws_size,
                              hipStream_t stream)
{
    const float* x     = (const float*)d_in[0];
    const float* W_in  = (const float*)d_in[1];
    const float* b_in  = (const float*)d_in[2];
    const float* W_out = (const float*)d_in[3];
    const float* b_out = (const float*)d_in[4];
    float* out = (float*)d_out;
    float* ws  = (float*)d_ws;   // needs B*NTILES*4 = 2 KB

    dim3 grid(NTILES, Bn);       // 8 x 64 = 512 workgroups, 8 waves each
    alif_fused<<<grid, 256, 0, stream>>>(x, W_in, b_in, W_out, ws);
    alif_finish<<<1, 64, 0, stream>>>(ws, b_out, out);
}